// TemporalTransformerBlockWithMemory_52621939311205
// MI455X (gfx1250) — compile-verified
//
#include <hip/hip_runtime.h>

// ---------------------------------------------------------------------------
// TemporalTransformerBlockWithMemory for MI455X (gfx1250), fp32 WMMA path.
// Memory-bound problem (~0.36 GB HBM traffic @ 23.3 TB/s ~= 15us floor), so
// fp32 V_WMMA_F32_16X16X4_F32 is used everywhere: exact-precision vs the f32
// reference and still far below the bandwidth ceiling.
// ---------------------------------------------------------------------------

typedef __attribute__((ext_vector_type(2))) float v2f;
typedef __attribute__((ext_vector_type(4))) float v4f;
typedef __attribute__((ext_vector_type(8))) float v8f;

#define HIDDEN_  1024
#define NHEAD_   16
#define HDIM_    64
#define LTOK_    1024
#define NTOK_    4096
#define KSTRIDE_ 68   // padded LDS row stride (floats) for 64-wide rows
#define PSTRIDE_ 18   // padded stride for 16x16 P scratch (even -> b64 aligned)

__device__ __forceinline__ v8f wmma4(v2f a, v2f b, v8f c) {
  // D = A(16x4 f32) * B(4x16 f32) + C(16x16 f32)
  return __builtin_amdgcn_wmma_f32_16x16x4_f32(
      /*neg_a=*/false, a, /*neg_b=*/false, b,
      /*c_mod=*/(short)0, c, /*reuse_a=*/false, /*reuse_b=*/false);
}

__device__ __forceinline__ float half16_sum(float v) {
  // sum across the 16-lane half-wave this lane belongs to
  v += __shfl_xor(v, 1, 16);
  v += __shfl_xor(v, 2, 16);
  v += __shfl_xor(v, 4, 16);
  v += __shfl_xor(v, 8, 16);
  return v;
}

// ---------------------------------------------------------------------------
// Attention over memory[6]: per (head, 64-query tile) block, loop groups g=0..3,
// stage K/V (1024x64) in LDS, streaming softmax (scores bounded -> no running
// max needed in fp32), accumulate the group-mean in registers.
// ---------------------------------------------------------------------------
extern "C" __global__ void __launch_bounds__(128)
attn_kernel(const float* __restrict__ mem6, float* __restrict__ att)
{
  extern __shared__ float smem[];
  const int h    = blockIdx.x;        // head 0..15
  const int qt   = blockIdx.y;        // query tile 0..15 (64 queries each)
  const int tid  = threadIdx.x;
  const int wave = tid >> 5;
  const int lane = tid & 31;
  const int hl   = lane & 15;         // lane within half
  const int hi   = lane >> 4;         // which half (0/1)

  float* Ksh = smem;                                    // [1024][KSTRIDE_]
  float* Psh = smem + LTOK_ * KSTRIDE_ + wave * (16 * PSTRIDE_);

  v8f attacc[4] = {};                                   // mean over groups

  for (int g = 0; g < 4; ++g) {
    __syncthreads();
    // stage K for group g, head h: token l -> frame token n
    for (int i = tid; i < LTOK_ * 16; i += 128) {
      int l  = i >> 4;
      int c4 = (i & 15) << 2;
      int r  = ((g >> 1) << 5) + (l >> 5);
      int cc = ((g & 1) << 5) + (l & 31);
      int n  = (r << 6) + cc;
      *(v4f*)&Ksh[l * KSTRIDE_ + c4] =
          *(const v4f*)&mem6[(size_t)n * HIDDEN_ + h * HDIM_ + c4];
    }
    __syncthreads();

    // Q tile (16 rows per wave) in A-layout: lane m=hl, k = 4c + 2*hi + {0,1}
    const int q0 = qt * 64 + wave * 16 + hl;
    v2f qa[16];
    #pragma unroll
    for (int c = 0; c < 16; ++c)
      qa[c] = *(const v2f*)&Ksh[q0 * KSTRIDE_ + 4 * c + 2 * hi];

    v8f o0 = {}, o1 = {}, o2 = {}, o3 = {};
    float ssum[8] = {0.f, 0.f, 0.f, 0.f, 0.f, 0.f, 0.f, 0.f};

    for (int kt = 0; kt < 64; ++kt) {
      // S = Q * K^T (16x16), chained over 16 k-chunks of 4 dims
      v8f s = {};
      const float* kb = &Ksh[(kt * 16 + hl) * KSTRIDE_ + 2 * hi];
      #pragma unroll
      for (int c = 0; c < 16; ++c) {
        v2f b = *(const v2f*)&kb[4 * c];
        s = wmma4(qa[c], b, s);
      }
      // P = exp(S/8); row sums; transpose P to A-layout via LDS scratch
      #pragma unroll
      for (int j = 0; j < 8; ++j) {
        float p = __expf(s[j] * 0.125f);
        ssum[j] += half16_sum(p);
        Psh[(j + 8 * hi) * PSTRIDE_ + hl] = p;          // row-major P
      }
      // O += P * V  (M=16 q, K=16 keys -> 4 chunks, N=64 dims -> 4 tiles)
      #pragma unroll
      for (int c = 0; c < 4; ++c) {
        v2f pa = *(const v2f*)&Psh[hl * PSTRIDE_ + 4 * c + 2 * hi];
        const float* vr = &Ksh[(kt * 16 + 4 * c + 2 * hi) * KSTRIDE_ + hl];
        v2f b;
        b.x = vr[0];  b.y = vr[KSTRIDE_];       o0 = wmma4(pa, b, o0);
        b.x = vr[16]; b.y = vr[KSTRIDE_ + 16];  o1 = wmma4(pa, b, o1);
        b.x = vr[32]; b.y = vr[KSTRIDE_ + 32];  o2 = wmma4(pa, b, o2);
        b.x = vr[48]; b.y = vr[KSTRIDE_ + 48];  o3 = wmma4(pa, b, o3);
      }
    }
    // normalize and accumulate group mean
    #pragma unroll
    for (int j = 0; j < 8; ++j) {
      float inv = 1.0f / ssum[j];
      attacc[0][j] += o0[j] * inv;
      attacc[1][j] += o1[j] * inv;
      attacc[2][j] += o2[j] * inv;
      attacc[3][j] += o3[j] * inv;
    }
  }

  #pragma unroll
  for (int j = 0; j < 8; ++j) {
    int q = qt * 64 + wave * 16 + j + 8 * hi;           // C/D row
    float* dst = &att[(size_t)q * HIDDEN_ + h * HDIM_ + hl];
    dst[0]  = attacc[0][j] * 0.25f;
    dst[16] = attacc[1][j] * 0.25f;
    dst[32] = attacc[2][j] * 0.25f;
    dst[48] = attacc[3][j] * 0.25f;
  }
}

// ---------------------------------------------------------------------------
// x_updated = x + att[n mod 1024]; LayerNorm -> y. One block per token.
// ---------------------------------------------------------------------------
extern "C" __global__ void __launch_bounds__(256)
ln_kernel(const float* __restrict__ x, const float* __restrict__ att,
          const float* __restrict__ gamma, const float* __restrict__ beta,
          float* __restrict__ xu, float* __restrict__ y)
{
  const int n   = blockIdx.x;
  const int tid = threadIdx.x;
  __shared__ float rs[8], rs2[8], stats[2];

  float v[4], s = 0.f, s2 = 0.f;
  #pragma unroll
  for (int i = 0; i < 4; ++i) {
    int c = tid + (i << 8);
    float val = x[(size_t)n * HIDDEN_ + c] +
                att[(size_t)(n & (LTOK_ - 1)) * HIDDEN_ + c];
    v[i] = val; s += val; s2 += val * val;
  }
  #pragma unroll
  for (int m = 16; m >= 1; m >>= 1) {
    s  += __shfl_xor(s,  m, 32);
    s2 += __shfl_xor(s2, m, 32);
  }
  if ((tid & 31) == 0) { rs[tid >> 5] = s; rs2[tid >> 5] = s2; }
  __syncthreads();
  if (tid == 0) {
    float a = 0.f, b = 0.f;
    for (int w = 0; w < 8; ++w) { a += rs[w]; b += rs2[w]; }
    float mu  = a * (1.0f / HIDDEN_);
    float var = b * (1.0f / HIDDEN_) - mu * mu;
    stats[0] = mu;
    stats[1] = rsqrtf(var + 1e-5f);
  }
  __syncthreads();
  float mu = stats[0], rstd = stats[1];
  #pragma unroll
  for (int i = 0; i < 4; ++i) {
    int c = tid + (i << 8);
    xu[(size_t)n * HIDDEN_ + c] = v[i];
    y [(size_t)n * HIDDEN_ + c] = (v[i] - mu) * rstd * gamma[c] + beta[c];
  }
}

// ---------------------------------------------------------------------------
// C = [relu](A*B + bias) [+ resid].  64x64 tile / block, 4 waves, fp32 WMMA.
// ---------------------------------------------------------------------------
#define GTM 64
#define GTK 64

extern "C" __global__ void __launch_bounds__(128)
gemm_kernel(const float* __restrict__ A, const float* __restrict__ B,
            const float* __restrict__ bias, const float* __restrict__ resid,
            float* __restrict__ C, int M, int N, int K, int relu)
{
  __shared__ float Ash[GTM * KSTRIDE_];
  __shared__ float Bsh[GTK * KSTRIDE_];
  const int bm = blockIdx.x, bn = blockIdx.y;
  const int tid  = threadIdx.x;
  const int wave = tid >> 5, lane = tid & 31;
  const int hl = lane & 15, hi = lane >> 4;

  v8f acc[4] = {};

  for (int kb = 0; kb < K; kb += GTK) {
    for (int i = tid; i < GTM * 16; i += 128) {
      int r = i >> 4, c4 = (i & 15) << 2;
      *(v4f*)&Ash[r * KSTRIDE_ + c4] =
          *(const v4f*)&A[(size_t)(bm * GTM + r) * K + kb + c4];
      *(v4f*)&Bsh[r * KSTRIDE_ + c4] =
          *(const v4f*)&B[(size_t)(kb + r) * N + bn * 64 + c4];
    }
    __syncthreads();
    #pragma unroll
    for (int c = 0; c < 16; ++c) {
      v2f a = *(const v2f*)&Ash[(wave * 16 + hl) * KSTRIDE_ + 4 * c + 2 * hi];
      const float* bp = &Bsh[(4 * c + 2 * hi) * KSTRIDE_ + hl];
      v2f b;
      b.x = bp[0];  b.y = bp[KSTRIDE_];       acc[0] = wmma4(a, b, acc[0]);
      b.x = bp[16]; b.y = bp[KSTRIDE_ + 16];  acc[1] = wmma4(a, b, acc[1]);
      b.x = bp[32]; b.y = bp[KSTRIDE_ + 32];  acc[2] = wmma4(a, b, acc[2]);
      b.x = bp[48]; b.y = bp[KSTRIDE_ + 48];  acc[3] = wmma4(a, b, acc[3]);
    }
    __syncthreads();
  }

  #pragma unroll
  for (int j = 0; j < 8; ++j) {
    int grow = bm * GTM + wave * 16 + j + 8 * hi;
    #pragma unroll
    for (int nt = 0; nt < 4; ++nt) {
      int col = bn * 64 + nt * 16 + hl;
      float v = acc[nt][j] + bias[col];
      if (relu)  v = fmaxf(v, 0.f);
      if (resid) v += resid[(size_t)grow * N + col];
      C[(size_t)grow * N + col] = v;
    }
  }
}

// ---------------------------------------------------------------------------
extern "C" void kernel_launch(void* const* d_in, const int* in_sizes, int n_in,
                              void* d_out, int out_size, void* d_ws, size_t ws_size,
                              hipStream_t stream)
{
  (void)in_sizes; (void)n_in; (void)out_size; (void)ws_size;
  const float* x      = (const float*)d_in[0];
  const float* memory = (const float*)d_in[1];
  const float* gamma  = (const float*)d_in[2];
  const float* beta   = (const float*)d_in[3];
  const float* w1     = (const float*)d_in[4];
  const float* b1     = (const float*)d_in[5];
  const float* w2     = (const float*)d_in[6];
  const float* b2     = (const float*)d_in[7];

  float* x_final = (float*)d_out;
  float* new_mem = x_final + (size_t)NTOK_ * HIDDEN_;

  // workspace: att 4MB | xu 16MB | y 16MB | H 64MB  (= 100MB)
  float* att = (float*)d_ws;
  float* xu  = att + (size_t)LTOK_ * HIDDEN_;
  float* yb  = xu  + (size_t)NTOK_ * HIDDEN_;
  float* hb  = yb  + (size_t)NTOK_ * HIDDEN_;

  // new_memory = [memory[1:7], x]  (pure DMA shift)
  hipMemcpyAsync(new_mem, memory + (size_t)NTOK_ * HIDDEN_,
                 (size_t)6 * NTOK_ * HIDDEN_ * sizeof(float),
                 hipMemcpyDeviceToDevice, stream);
  hipMemcpyAsync(new_mem + (size_t)6 * NTOK_ * HIDDEN_, x,
                 (size_t)NTOK_ * HIDDEN_ * sizeof(float),
                 hipMemcpyDeviceToDevice, stream);

  const float* mem6 = memory + (size_t)6 * NTOK_ * HIDDEN_;  // x_cat[-1]
  size_t smem = ((size_t)LTOK_ * KSTRIDE_ + 4 * 16 * PSTRIDE_) * sizeof(float);
  attn_kernel<<<dim3(NHEAD_, LTOK_ / 64), 128, smem, stream>>>(mem6, att);

  ln_kernel<<<NTOK_, 256, 0, stream>>>(x, att, gamma, beta, xu, yb);

  gemm_kernel<<<dim3(NTOK_ / 64, 4096 / 64), 128, 0, stream>>>(
      yb, w1, b1, nullptr, hb, NTOK_, 4096, HIDDEN_, 1);
  gemm_kernel<<<dim3(NTOK_ / 64, HIDDEN_ / 64), 128, 0, stream>>>(
      hb, w2, b2, xu, x_final, NTOK_, HIDDEN_, 4096, 0);
}